// myGAT_40218073759959
// MI455X (gfx1250) — compile-verified
//
#include <hip/hip_runtime.h>
#include <hip/hip_bf16.h>

// ---------------------------------------------------------------------------
// Types for CDNA5 WMMA (gfx1250, wave32)
// ---------------------------------------------------------------------------
typedef __attribute__((ext_vector_type(2))) float v2f;
typedef __attribute__((ext_vector_type(8))) float v8f;

#define NEG_SLOPE 0.2f

// Monotone mapping: float (finite) -> int such that int compare == float compare.
__device__ __forceinline__ int f32_to_ordered(float f) {
    int i = __float_as_int(f);
    return i >= 0 ? i : (i ^ 0x7fffffff);
}
__device__ __forceinline__ float ordered_to_f32(int i) {
    return __int_as_float(i >= 0 ? i : (i ^ 0x7fffffff));
}

// ---------------------------------------------------------------------------
// K0: init out = 0, m = INT_MIN (== -inf under ordered mapping), s = 0
// ---------------------------------------------------------------------------
__global__ void k_init(float* __restrict__ out, int* __restrict__ mI,
                       float* __restrict__ ssum, long n_out, int n4) {
    long i = (long)blockIdx.x * blockDim.x + threadIdx.x;
    if (i < n_out) out[i] = 0.0f;
    if (i < n4) { mI[i] = (int)0x80000000; ssum[i] = 0.0f; }
}

// ---------------------------------------------------------------------------
// K1: feat = x @ W via V_WMMA_F32_16X16X4_F32.
//   x: [N,64], W: [64,128], feat: [N,128].
//   256 threads = 8 waves; each wave computes a 16x128 row tile.
//   W staged in LDS K-pair-packed (sWp[p][col] = {W[2p][col], W[2p+1][col]})
//   so both A and B fragments are single aligned ds_load_b64 ops.
// ---------------------------------------------------------------------------
__global__ void k_gemm_feat(const float* __restrict__ x, const float* __restrict__ W,
                            float* __restrict__ feat, int n_nodes) {
    __shared__ float2 sWp[32 * 128];    // 32 KB: [k/2][col] -> {W[k][col], W[k+1][col]}
    __shared__ float  sX[8][16 * 64];   // 32 KB: per-wave 16x64 x tiles

    const int tid  = threadIdx.x;
    const int w    = tid >> 5;          // wave in block (0..7)
    const int lane = tid & 31;
    const int half = lane >> 4;         // 0: K pair {0,1}, 1: K pair {2,3}
    const int l16  = lane & 15;

    // Stage W K-pair-packed: 4096 float2, 256 threads x 16.
    // Per iteration: lanes read contiguous 128B runs of rows 2p and 2p+1.
    for (int i = tid; i < 32 * 128; i += 256) {
        int p = i >> 7, col = i & 127;
        float2 v;
        v.x = W[(2 * p + 0) * 128 + col];
        v.y = W[(2 * p + 1) * 128 + col];
        sWp[i] = v;
    }

    // Stage this wave's 16x64 x tile (row-clamped for tail waves)
    const int wave_row0 = (blockIdx.x * 8 + w) * 16;
    for (int i = lane * 4; i < 16 * 64; i += 32 * 4) {
        int r = i >> 6, c = i & 63;
        int gr = wave_row0 + r;
        if (gr >= n_nodes) gr = n_nodes - 1;
        *(float4*)&sX[w][i] = *(const float4*)&x[(size_t)gr * 64 + c];
    }
    __syncthreads();

    if (wave_row0 >= n_nodes) return;   // uniform per wave; after barrier

    const float* xt = sX[w];
    v8f acc[8] = {};

    for (int k = 0; k < 64; k += 4) {
        // A fragment: lane holds row M=l16, K = k + half*2 + {0,1} (8B aligned)
        v2f a = *(const v2f*)&xt[l16 * 64 + k + half * 2];
        const int bp = ((k + half * 2) >> 1) * 128;
#pragma unroll
        for (int t = 0; t < 8; ++t) {
            // B fragment: single ds_load_b64 of the packed K pair for col N
            v2f b = *(const v2f*)&sWp[bp + t * 16 + l16];
            acc[t] = __builtin_amdgcn_wmma_f32_16x16x4_f32(
                false, a, false, b, (short)0, acc[t], false, false);
        }
    }

    // C/D layout: VGPR r, lanes 0-15 -> M=r, lanes 16-31 -> M=r+8; N = l16
#pragma unroll
    for (int t = 0; t < 8; ++t) {
#pragma unroll
        for (int r = 0; r < 8; ++r) {
            int row = wave_row0 + r + half * 8;
            if (row < n_nodes)
                feat[(size_t)row * 128 + t * 16 + l16] = acc[t][r];
        }
    }
}

// ---------------------------------------------------------------------------
// K2: el[n,h] = feat[n,h,:].attn_l[h];  er likewise. One thread per (n,h).
// ---------------------------------------------------------------------------
__global__ void k_node_attn(const float* __restrict__ feat,
                            const float* __restrict__ attn_l,
                            const float* __restrict__ attn_r,
                            float* __restrict__ el, float* __restrict__ er, int n4) {
    int i = blockIdx.x * blockDim.x + threadIdx.x;
    if (i >= n4) return;
    int h = i & 3;
    const float* f  = feat + (size_t)(i >> 2) * 128 + h * 32;
    const float* al = attn_l + h * 32;
    const float* ar = attn_r + h * 32;
    float sl = 0.f, sr = 0.f;
#pragma unroll
    for (int d = 0; d < 32; d += 4) {
        float4 fv = *(const float4*)(f + d);
        float4 av = *(const float4*)(al + d);
        float4 bv = *(const float4*)(ar + d);
        sl += fv.x * av.x + fv.y * av.y + fv.z * av.z + fv.w * av.w;
        sr += fv.x * bv.x + fv.y * bv.y + fv.z * bv.z + fv.w * bv.w;
    }
    el[i] = sl;
    er[i] = sr;
}

// ---------------------------------------------------------------------------
// K3: e_att_type[t,h] = sum_f (edge_emb[t,:] @ W_e[:, h*64+f]) * attn_e[h,f]
//     T*H = 32 outputs, trivially small.
// ---------------------------------------------------------------------------
__global__ void k_etype_attn(const float* __restrict__ edge_emb,
                             const float* __restrict__ W_e,
                             const float* __restrict__ attn_e,
                             float* __restrict__ eatt) {
    int i = threadIdx.x;
    if (i >= 32) return;
    int t = i >> 2, h = i & 3;
    float r = 0.f;
    for (int f = 0; f < 64; ++f) {
        float s = 0.f;
        for (int k = 0; k < 64; ++k)
            s += edge_emb[t * 64 + k] * W_e[k * 256 + h * 64 + f];
        r += s * attn_e[h * 64 + f];
    }
    eatt[i] = r;
}

// ---------------------------------------------------------------------------
// K4: per-edge raw score + leaky_relu; atomicMax into per-(dst,h) running max.
// ---------------------------------------------------------------------------
__global__ void k_edge_score(const float* __restrict__ el, const float* __restrict__ er,
                             const float* __restrict__ eatt,
                             const int* __restrict__ src, const int* __restrict__ dst,
                             const int* __restrict__ etype,
                             float* __restrict__ esc, int* __restrict__ mI, int n_edges) {
    int e = blockIdx.x * blockDim.x + threadIdx.x;
    if (e >= n_edges) return;
    int sv = src[e], dv = dst[e], tv = etype[e];
    float4 a = *(const float4*)(el + (size_t)sv * 4);
    float4 b = *(const float4*)(er + (size_t)dv * 4);
    float4 c = *(const float4*)(eatt + (size_t)tv * 4);
    float4 v;
    v.x = a.x + b.x + c.x; v.x = v.x >= 0.f ? v.x : NEG_SLOPE * v.x;
    v.y = a.y + b.y + c.y; v.y = v.y >= 0.f ? v.y : NEG_SLOPE * v.y;
    v.z = a.z + b.z + c.z; v.z = v.z >= 0.f ? v.z : NEG_SLOPE * v.z;
    v.w = a.w + b.w + c.w; v.w = v.w >= 0.f ? v.w : NEG_SLOPE * v.w;
    *(float4*)(esc + (size_t)e * 4) = v;
    int* mp = mI + (size_t)dv * 4;
    atomicMax(mp + 0, f32_to_ordered(v.x));
    atomicMax(mp + 1, f32_to_ordered(v.y));
    atomicMax(mp + 2, f32_to_ordered(v.z));
    atomicMax(mp + 3, f32_to_ordered(v.w));
}

// ---------------------------------------------------------------------------
// K5: ex = exp(e - m[dst]); esc <- ex; atomicAdd into per-(dst,h) sum.
// ---------------------------------------------------------------------------
__global__ void k_softmax_num(float* __restrict__ esc, const int* __restrict__ dst,
                              const int* __restrict__ mI, float* __restrict__ ssum,
                              int n_edges) {
    int e = blockIdx.x * blockDim.x + threadIdx.x;
    if (e >= n_edges) return;
    int dv = dst[e];
    const int* mp = mI + (size_t)dv * 4;
    float4 v = *(const float4*)(esc + (size_t)e * 4);
    float4 ex;
    ex.x = expf(v.x - ordered_to_f32(mp[0]));
    ex.y = expf(v.y - ordered_to_f32(mp[1]));
    ex.z = expf(v.z - ordered_to_f32(mp[2]));
    ex.w = expf(v.w - ordered_to_f32(mp[3]));
    *(float4*)(esc + (size_t)e * 4) = ex;
    float* sp = ssum + (size_t)dv * 4;
    atomicAdd(sp + 0, ex.x);
    atomicAdd(sp + 1, ex.y);
    atomicAdd(sp + 2, ex.z);
    atomicAdd(sp + 3, ex.w);
}

// ---------------------------------------------------------------------------
// K6: aggregation. 32 lanes per edge; lane covers 4 consecutive channels of
//     the 128-wide message. alpha = ex/s[dst]; out[dst] += alpha*feat[src].
//     feat (51MB) and out (51MB) are L2-resident (192MB L2).
// ---------------------------------------------------------------------------
__global__ void k_aggregate(const float* __restrict__ feat, const float* __restrict__ esc,
                            const float* __restrict__ ssum,
                            const int* __restrict__ src, const int* __restrict__ dst,
                            float* __restrict__ out, int n_edges) {
    long gt = (long)blockIdx.x * blockDim.x + threadIdx.x;
    int e = (int)(gt >> 5);
    int lane = (int)(gt & 31);
    if (e >= n_edges) return;
    int sv = src[e], dv = dst[e];
    int h = lane >> 3;                                       // 0..3 (32 ch per head)
    float alpha = esc[(size_t)e * 4 + h] / ssum[(size_t)dv * 4 + h];
    float4 f4 = *(const float4*)(feat + (size_t)sv * 128 + lane * 4);
    float* op = out + (size_t)dv * 128 + lane * 4;
    atomicAdd(op + 0, alpha * f4.x);
    atomicAdd(op + 1, alpha * f4.y);
    atomicAdd(op + 2, alpha * f4.z);
    atomicAdd(op + 3, alpha * f4.w);
}

// ---------------------------------------------------------------------------
// K7: out = elu(out + bias)
// ---------------------------------------------------------------------------
__global__ void k_finalize(float* __restrict__ out, const float* __restrict__ bias,
                           long total) {
    long i = (long)blockIdx.x * blockDim.x + threadIdx.x;
    if (i >= total) return;
    float v = out[i] + bias[i & 127];
    out[i] = v > 0.f ? v : expm1f(v);
}

// ---------------------------------------------------------------------------
static inline int cdiv(long a, long b) { return (int)((a + b - 1) / b); }

extern "C" void kernel_launch(void* const* d_in, const int* in_sizes, int n_in,
                              void* d_out, int out_size, void* d_ws, size_t ws_size,
                              hipStream_t stream) {
    const float* x        = (const float*)d_in[0];
    const float* W        = (const float*)d_in[1];
    const float* attn_l   = (const float*)d_in[2];
    const float* attn_r   = (const float*)d_in[3];
    const float* edge_emb = (const float*)d_in[4];
    const float* W_e      = (const float*)d_in[5];
    const float* attn_e   = (const float*)d_in[6];
    const float* bias     = (const float*)d_in[7];
    const int*   src      = (const int*)d_in[8];
    const int*   dst      = (const int*)d_in[9];
    const int*   etype    = (const int*)d_in[10];
    float* out = (float*)d_out;

    const int N = in_sizes[0] / 64;   // 100000
    const int E = in_sizes[8];        // 1600000

    // workspace layout (all 16B-aligned)
    float* feat = (float*)d_ws;                       // N*128
    float* el   = feat + (size_t)N * 128;             // N*4
    float* er   = el + (size_t)N * 4;                 // N*4
    float* eatt = er + (size_t)N * 4;                 // 32
    int*   mI   = (int*)(eatt + 32);                  // N*4
    float* ssum = (float*)(mI + (size_t)N * 4);       // N*4
    float* esc  = ssum + (size_t)N * 4;               // E*4

    const long n_out = (long)N * 128;

    k_init<<<cdiv(n_out, 256), 256, 0, stream>>>(out, mI, ssum, n_out, N * 4);
    k_gemm_feat<<<cdiv(N, 128), 256, 0, stream>>>(x, W, feat, N);
    k_node_attn<<<cdiv((long)N * 4, 256), 256, 0, stream>>>(feat, attn_l, attn_r, el, er, N * 4);
    k_etype_attn<<<1, 32, 0, stream>>>(edge_emb, W_e, attn_e, eatt);
    k_edge_score<<<cdiv(E, 256), 256, 0, stream>>>(el, er, eatt, src, dst, etype, esc, mI, E);
    k_softmax_num<<<cdiv(E, 256), 256, 0, stream>>>(esc, dst, mI, ssum, E);
    k_aggregate<<<cdiv((long)E * 32, 256), 256, 0, stream>>>(feat, esc, ssum, src, dst, out, E);
    k_finalize<<<cdiv(n_out, 256), 256, 0, stream>>>(out, bias, n_out);
}